// ModelFedQuitLogitDynamicAlternative_46368466927937
// MI455X (gfx1250) — compile-verified
//
#include <hip/hip_runtime.h>

// Problem constants (from reference): B=4096, C=32000, K=10, TAU=1
#define CC   32000
#define KK   10
#define TPB  256
#define NW   (TPB / 32)   // 8 waves per block (wave32)
#define NV4  (CC / 4)     // 8000 float4 per row

typedef __attribute__((ext_vector_type(4))) float v4f;

// Monotone mapping float -> unsigned for packed max-reduction
__device__ __forceinline__ unsigned fkey(float x) {
    unsigned u = __float_as_uint(x);
    return (u & 0x80000000u) ? ~u : (u | 0x80000000u);
}
__device__ __forceinline__ float finv(unsigned k) {
    unsigned u = (k & 0x80000000u) ? (k & 0x7fffffffu) : ~k;
    return __uint_as_float(u);
}

// Wave32 butterfly reductions (all lanes end with the result; no barriers)
__device__ __forceinline__ float wave_min_f32(float v) {
#pragma unroll
    for (int m = 16; m >= 1; m >>= 1) v = fminf(v, __shfl_xor(v, m, 32));
    return v;
}
__device__ __forceinline__ float wave_sum_f32(float v) {
#pragma unroll
    for (int m = 16; m >= 1; m >>= 1) v += __shfl_xor(v, m, 32);
    return v;
}
__device__ __forceinline__ unsigned long long wave_max_u64(unsigned long long v) {
#pragma unroll
    for (int m = 16; m >= 1; m >>= 1) {
        unsigned lo  = (unsigned)v, hi = (unsigned)(v >> 32);
        unsigned olo = (unsigned)__shfl_xor((int)lo, m, 32);
        unsigned ohi = (unsigned)__shfl_xor((int)hi, m, 32);
        unsigned long long o = ((unsigned long long)ohi << 32) | olo;
        v = (o > v) ? o : v;
    }
    return v;
}

__global__ __launch_bounds__(TPB)
void fedquit_virtual_teacher_kernel(const float* __restrict__ zg,
                                    const int*   __restrict__ y,
                                    float*       __restrict__ out)
{
    // 128000 B row buffer + ~200 B reduction scratch -> 2 workgroups / 320 KB WGP
    __shared__ __align__(16) float zrow[CC];
    __shared__ unsigned long long  cred8[2][NW];  // per-round parity double buffer
    __shared__ float               fred8[NW];

    const int row  = blockIdx.x;
    const int tid  = threadIdx.x;
    const int wave = tid >> 5;
    const int lane = tid & 31;
    const int yi   = y[row];
    const float* __restrict__ zr   = zg  + (size_t)row * CC;
    float*       __restrict__ orow = out + (size_t)row * CC;

    // ---------------- Stage row HBM -> LDS with CDNA5 async copies ----------
    {
        // low 32 bits of a generic pointer to __shared__ == LDS byte address
        unsigned ldsbase = (unsigned)(unsigned long long)(&zrow[0]);
        for (int j = tid; j < NV4; j += TPB) {
            unsigned           lds = ldsbase + (unsigned)(j * 16);
            unsigned long long ga  = (unsigned long long)(const void*)(zr + j * 4);
            asm volatile("global_load_async_to_lds_b128 %0, %1, off"
                         :: "v"(lds), "v"(ga) : "memory");
        }
    }
    asm volatile("s_wait_asynccnt 0" ::: "memory");
    __syncthreads();

    // ---------------- Pass A: row min + per-thread top-10 of non-true -------
    float lmin = 3.4e38f;
    float tv[KK];
#pragma unroll
    for (int k = 0; k < KK; ++k) tv[k] = -1e30f;   // descending sorted list

    for (int j = tid; j < NV4; j += TPB) {
        v4f z4 = *reinterpret_cast<const v4f*>(&zrow[j * 4]);
        const int i0 = j * 4;
#pragma unroll
        for (int c = 0; c < 4; ++c) {
            float x = z4[c];
            lmin = fminf(lmin, x);
            if (((i0 + c) != yi) && (x > tv[KK - 1])) {
                tv[KK - 1] = x;                     // bubble up (rare path)
#pragma unroll
                for (int k = KK - 1; k > 0; --k) {
                    float a = tv[k], b = tv[k - 1];
                    tv[k]     = fminf(a, b);
                    tv[k - 1] = fmaxf(a, b);
                }
            }
        }
    }

    // block min-reduce: wave butterfly + one 8-entry LDS hop
    lmin = wave_min_f32(lmin);
    if (lane == 0) fred8[wave] = lmin;
    __syncthreads();
    float vmin = fred8[0];
#pragma unroll
    for (int w = 1; w < NW; ++w) vmin = fminf(vmin, fred8[w]);

    // 10 rounds of block arg-max over per-thread sorted heads -> global top-10
    float topv[KK];
    int pos = 0;
#pragma unroll
    for (int r = 0; r < KK; ++r) {
        float head = (pos < KK) ? tv[pos] : -1e30f;
        unsigned long long offer =
            ((unsigned long long)fkey(head) << 32) | (unsigned)tid;
        offer = wave_max_u64(offer);                   // wave-local argmax
        if (lane == 0) cred8[r & 1][wave] = offer;
        __syncthreads();                               // one barrier per round
        unsigned long long win = cred8[r & 1][0];
#pragma unroll
        for (int w = 1; w < NW; ++w) {
            unsigned long long o = cred8[r & 1][w];
            win = (o > win) ? o : win;
        }
        topv[r] = finv((unsigned)(win >> 32));
        if ((unsigned)(win & 0xffffffffu) == (unsigned)tid) ++pos;  // pop head
    }
    const float tthr = topv[KK - 1];   // 10th largest non-true logit

    // ---------------- Pass B: sum exp(z - t) over "rest" --------------------
    float s_l = 0.f;
    for (int j = tid; j < NV4; j += TPB) {
        v4f z4 = *reinterpret_cast<const v4f*>(&zrow[j * 4]);
        const int i0 = j * 4;
#pragma unroll
        for (int c = 0; c < 4; ++c) {
            float x = z4[c];
            if (((i0 + c) != yi) && (x < tthr)) s_l += __expf(x - tthr);
        }
    }
    s_l = wave_sum_f32(s_l);
    __syncthreads();                    // protect fred8 reuse (min phase done)
    if (lane == 0) fred8[wave] = s_l;
    __syncthreads();
    float Srest = fred8[0];
#pragma unroll
    for (int w = 1; w < NW; ++w) Srest += fred8[w];

    // ---------------- Per-row scalars (all threads, redundant) --------------
    const float MR     = (float)(CC - 1 - KK);        // 31989
    const float alphaK = tthr + __logf(Srest) - __logf(MR);
    const float m2     = fmaxf(topv[0], fmaxf(alphaK, vmin));
    float D = __expf(alphaK - m2) * MR + __expf(vmin - m2);
#pragma unroll
    for (int k = 0; k < KK; ++k) D += __expf(topv[k] - m2);
    const float invD   = 1.0f / D;
    const float p_rest = __expf(alphaK - m2) * invD;
    const float p_true = __expf(vmin   - m2) * invD;

    // ---------------- Pass C: write p_virt (non-temporal b128 stores) -------
    for (int j = tid; j < NV4; j += TPB) {
        v4f z4 = *reinterpret_cast<const v4f*>(&zrow[j * 4]);
        const int i0 = j * 4;
        v4f ov;
#pragma unroll
        for (int c = 0; c < 4; ++c) {
            float x = z4[c];
            float p = (x >= tthr) ? (__expf(x - m2) * invD) : p_rest;
            if ((i0 + c) == yi) p = p_true;
            ov[c] = p;
        }
        __builtin_nontemporal_store(ov, reinterpret_cast<v4f*>(&orow[j * 4]));
    }
}

extern "C" void kernel_launch(void* const* d_in, const int* in_sizes, int n_in,
                              void* d_out, int out_size, void* d_ws, size_t ws_size,
                              hipStream_t stream) {
    const float* zg = (const float*)d_in[0];   // [B, C] fp32
    const int*   y  = (const int*)d_in[1];     // [B] int32
    float*       out = (float*)d_out;          // [B, C] fp32
    const int B = in_sizes[1];                 // 4096 rows, one block per row
    fedquit_virtual_teacher_kernel<<<B, TPB, 0, stream>>>(zg, y, out);
}